// PerSampleQuantileMinMaxNormalize_60464549593548
// MI455X (gfx1250) — compile-verified
//
#include <hip/hip_runtime.h>
#include <stdint.h>

// Problem geometry (from reference): x is (32, 7, 512, 512) float32.
#define B_DIM     32
#define C_DIM     7
#define SLICES    (B_DIM * C_DIM)      // 224
#define HW        262144               // 512*512 elements per slice
#define QTHREADS  1024                 // 32 waves (wave32)
#define NTILES    (HW / (QTHREADS * 4))// 64 tiles of 1024 float4

// 64K-bin histogram with one padding dword per 64 bins -> conflict-free
// stride-65 per-thread reduction under the 64-bank LDS.
#define HBINS     65536
#define HSLOTS    (HBINS + (HBINS >> 6))   // 66560 dwords = 260 KB
__device__ __forceinline__ int hidx(unsigned b) { return (int)(b + (b >> 6)); }

// ---- monotone key mapping: float -> unsigned preserving order ----
__device__ __forceinline__ unsigned orderKey(float f) {
    unsigned b = __float_as_uint(f);
    return (b & 0x80000000u) ? ~b : (b | 0x80000000u);
}
__device__ __forceinline__ float keyToFloat(unsigned u) {
    unsigned b = (u & 0x80000000u) ? (u ^ 0x80000000u) : ~u;
    return __uint_as_float(b);
}

// One workgroup per (b,c) slice. Exact MSB-first radix select (16+8+8 bits)
// for the four order statistics {2621, 2622, 259521, 259522} simultaneously.
__global__ __launch_bounds__(QTHREADS, 1)
void quantile_kernel(const float* __restrict__ x, float2* __restrict__ params) {
    __shared__ unsigned histA[HSLOTS];       // 260 KB: 16-bit first level
    __shared__ float4   stage[2][QTHREADS];  // 32 KB async double buffer
    __shared__ unsigned histR[4][257];       // per-rank conditioned hist
    __shared__ unsigned partial[QTHREADS];   // block-scan workspace
    __shared__ unsigned sPrefix[4];
    __shared__ unsigned sRank[4];
    __shared__ float    sVal[4];

    const int tid   = threadIdx.x;
    const int slice = blockIdx.x;
    const float* src = x + (size_t)slice * HW;

    for (int i = tid; i < HSLOTS; i += QTHREADS) histA[i] = 0u;
    __syncthreads();

    // ---------- Pass A: top 16 bits, async-staged through LDS ----------
    const unsigned ldsAddr0 = (unsigned)(uintptr_t)&stage[0][tid];
    const unsigned ldsAddr1 = (unsigned)(uintptr_t)&stage[1][tid];
    {
        unsigned voff = (unsigned)tid * 16u;
        asm volatile("global_load_async_to_lds_b128 %0, %1, %2"
                     :: "v"(ldsAddr0), "v"(voff), "s"(src) : "memory");
    }
    for (int it = 0; it < NTILES; ++it) {
        if (it + 1 < NTILES) {
            unsigned voff = (unsigned)(it + 1) * (QTHREADS * 16u) + (unsigned)tid * 16u;
            asm volatile("global_load_async_to_lds_b128 %0, %1, %2"
                         :: "v"((it & 1) ? ldsAddr0 : ldsAddr1), "v"(voff), "s"(src)
                         : "memory");
            asm volatile("s_wait_asynccnt 0x1" ::: "memory");
        } else {
            asm volatile("s_wait_asynccnt 0x0" ::: "memory");
        }
        float4 v = stage[it & 1][tid];   // each lane reads its own staged slot
        atomicAdd(&histA[hidx(orderKey(v.x) >> 16)], 1u);
        atomicAdd(&histA[hidx(orderKey(v.y) >> 16)], 1u);
        atomicAdd(&histA[hidx(orderKey(v.z) >> 16)], 1u);
        atomicAdd(&histA[hidx(orderKey(v.w) >> 16)], 1u);
    }
    __syncthreads();

    // ---------- locate the 4 ranks in the 64K-bin histogram ----------
    // per-thread partial sum over its 64 contiguous bins (stride-65 base:
    // bank = (tid*65 + j) & 63 = (tid + j) & 63 -> conflict-free)
    const int hbase = tid * 65;
    unsigned own = 0;
    #pragma unroll
    for (int j = 0; j < 64; ++j) own += histA[hbase + j];
    partial[tid] = own;
    __syncthreads();

    // inclusive Hillis-Steele scan over 1024 partials
    for (int off = 1; off < QTHREADS; off <<= 1) {
        unsigned v = (tid >= off) ? partial[tid - off] : 0u;
        __syncthreads();
        partial[tid] += v;
        __syncthreads();
    }
    const unsigned excl = partial[tid] - own;

    #pragma unroll
    for (int r = 0; r < 4; ++r) {
        const unsigned k = (r < 2) ? (2621u + (unsigned)r) : (259521u + (unsigned)(r - 2));
        if (k >= excl && k < excl + own) {       // exactly one owner thread per rank
            unsigned cum = excl;
            for (int j = 0; j < 64; ++j) {
                unsigned cnt = histA[hbase + j];
                if (k < cum + cnt) {
                    sPrefix[r] = (unsigned)(tid * 64 + j) << 16;
                    sRank[r]   = k - cum;
                    break;
                }
                cum += cnt;
            }
        }
    }
    __syncthreads();

    // ---------- Passes B,C: prefix-conditioned 8-bit refinement ----------
    const float4* src4 = (const float4*)src;
    #pragma unroll 1
    for (int p = 2; p <= 3; ++p) {
        const int shift   = 24 - 8 * p;   // 8, then 0
        const int shiftHi = 32 - 8 * p;   // compare top 16, then top 24 bits
        {
            unsigned* hR = &histR[0][0];
            for (int i = tid; i < 4 * 257; i += QTHREADS) hR[i] = 0u;
        }
        __syncthreads();
        const unsigned pr0 = sPrefix[0] >> shiftHi;
        const unsigned pr1 = sPrefix[1] >> shiftHi;
        const unsigned pr2 = sPrefix[2] >> shiftHi;
        const unsigned pr3 = sPrefix[3] >> shiftHi;

        for (int it = 0; it < NTILES; ++it) {
            float4 v = src4[it * QTHREADS + tid];
            #pragma unroll
            for (int e = 0; e < 4; ++e) {
                float f = (e == 0) ? v.x : (e == 1) ? v.y : (e == 2) ? v.z : v.w;
                unsigned u  = orderKey(f);
                unsigned hi = u >> shiftHi;
                unsigned dg = (u >> shift) & 255u;
                if (hi == pr0) atomicAdd(&histR[0][dg], 1u);
                if (hi == pr1) atomicAdd(&histR[1][dg], 1u);
                if (hi == pr2) atomicAdd(&histR[2][dg], 1u);
                if (hi == pr3) atomicAdd(&histR[3][dg], 1u);
            }
        }
        __syncthreads();
        if (tid < 4) {
            unsigned k = sRank[tid];
            unsigned cum = 0, digit = 0;
            for (int d = 0; d < 256; ++d) {
                unsigned cnt = histR[tid][d];
                if (k < cum + cnt) { digit = (unsigned)d; break; }
                cum += cnt;
            }
            sRank[tid]    = k - cum;
            sPrefix[tid] |= digit << shift;
        }
        __syncthreads();
    }

    // ---------- interpolate quantiles, emit per-slice affine ----------
    if (tid < 4) sVal[tid] = keyToFloat(sPrefix[tid]);
    __syncthreads();
    if (tid == 0) {
        const double posLo = 0.01 * (double)(HW - 1);     // 2621.43
        const double posHi = 0.99 * (double)(HW - 1);     // 259521.57
        const double fLo = posLo - 2621.0;
        const double fHi = posHi - 259521.0;
        double v0 = (double)sVal[0], v1 = (double)sVal[1];
        double v2 = (double)sVal[2], v3 = (double)sVal[3];
        double vmin = v0 + fLo * (v1 - v0);
        double vmax = v2 + fHi * (v3 - v2);
        double scale = 1.0 / (vmax - vmin + 1e-8);
        params[slice] = make_float2((float)scale, (float)(-vmin * scale));
    }
}

// Streaming normalize: out = clamp(x*a + b, 0, 1), float4-vectorized.
__global__ __launch_bounds__(256, 4)
void normalize_kernel(const float4* __restrict__ x4,
                      const float2* __restrict__ params,
                      float4* __restrict__ out4) {
    const int gid   = blockIdx.x * 256 + threadIdx.x;
    const int slice = blockIdx.x >> 8;        // 256 blocks (1024 elems each) per slice
    const float2 p  = params[slice];
    float4 v = x4[gid];
    v.x = fminf(fmaxf(fmaf(v.x, p.x, p.y), 0.0f), 1.0f);
    v.y = fminf(fmaxf(fmaf(v.y, p.x, p.y), 0.0f), 1.0f);
    v.z = fminf(fmaxf(fmaf(v.z, p.x, p.y), 0.0f), 1.0f);
    v.w = fminf(fmaxf(fmaf(v.w, p.x, p.y), 0.0f), 1.0f);
    out4[gid] = v;
}

extern "C" void kernel_launch(void* const* d_in, const int* in_sizes, int n_in,
                              void* d_out, int out_size, void* d_ws, size_t ws_size,
                              hipStream_t stream) {
    (void)in_sizes; (void)n_in; (void)ws_size;
    const float* x = (const float*)d_in[0];
    float2* params = (float2*)d_ws;           // 224 * 8 bytes of scratch

    quantile_kernel<<<SLICES, QTHREADS, 0, stream>>>(x, params);

    const int totalV4 = out_size / 4;         // 14,680,064 float4
    const int blocks  = totalV4 / 256;        // 57,344 blocks
    normalize_kernel<<<blocks, 256, 0, stream>>>((const float4*)x, params, (float4*)d_out);
}